// RealVirtualAttention_45535243272772
// MI455X (gfx1250) — compile-verified
//
#include <hip/hip_runtime.h>
#include <hip/hip_bf16.h>

#define VIRTUAL_Z 100

typedef __attribute__((ext_vector_type(2))) float v2f;
typedef __attribute__((ext_vector_type(8))) float v8f;

// ---------------------------------------------------------------------------
// Kernel 1: masked segment mean. batch[] is sorted, so one workgroup per
// graph binary-searches its node range and streams rows coalesced (thread t
// owns feature dim t). Writes stacked[g][m][d] means directly (no atomics).
// Dominant kernel: streams the full 600 MB feature matrix exactly once.
// ---------------------------------------------------------------------------
__global__ void seg_mean_kernel(const float* __restrict__ feat,   // [N, D]
                                const int* __restrict__ z,        // [N]
                                const int* __restrict__ batch,    // [N] sorted
                                float* __restrict__ stacked,      // [B, 2, D]
                                int N, int D) {
  const int g = blockIdx.x;

  // lower_bound(g)
  int lo = 0, hi = N;
  while (lo < hi) { int mid = (lo + hi) >> 1; if (batch[mid] < g) lo = mid + 1; else hi = mid; }
  const int start = lo;
  // upper_bound(g)
  hi = N;
  while (lo < hi) { int mid = (lo + hi) >> 1; if (batch[mid] <= g) lo = mid + 1; else hi = mid; }
  const int end = lo;

  const int d = threadIdx.x;
  if (d >= D) return;

  float accR = 0.f, accV = 0.f, cntR = 0.f, cntV = 0.f;
  for (int n = start; n < end; ++n) {
    const bool real = (z[n] != VIRTUAL_Z);                         // one request/wave
    const float v = __builtin_nontemporal_load(&feat[(long long)n * D + d]);
    if (real) { accR += v; cntR += 1.f; }
    else      { accV += v; cntV += 1.f; }
  }
  const long long base = (long long)g * 2 * D;
  stacked[base + d]     = accR / fmaxf(cntR, 1.f);
  stacked[base + D + d] = accV / fmaxf(cntV, 1.f);
}

// ---------------------------------------------------------------------------
// Kernel 2: w[row] = q^T tanh(W1^T x_row + b1) via V_WMMA_F32_16X16X4_F32.
// One wave handles a 16-row tile across all A=128 cols (8 N-tiles of 16).
// Main K-loop is fully in-bounds (no masking, no branches); a single tail
// step handles D%4 with clamped addresses + value selects (v_cndmask).
//   A frag (16x4 f32): lanes 0-15 -> K=k0,k0+1 ; lanes 16-31 -> K=k0+2,k0+3
//   B frag (4x16 f32): same K split; lane&15 selects column N.
//   C/D (16x16 f32):  vgpr r, lanes 0-15 -> M=r ; lanes 16-31 -> M=r+8.
// ---------------------------------------------------------------------------
__global__ void attn_score_kernel(const float* __restrict__ X,   // [R, D] stacked
                                  const float* __restrict__ W1,  // [D, A]
                                  const float* __restrict__ b1,  // [A]
                                  const float* __restrict__ q,   // [A]
                                  float* __restrict__ w,         // [R]
                                  int R, int D, int A) {
  const int lane   = threadIdx.x & 31;
  const int waveId = threadIdx.x >> 5;
  const int tile   = blockIdx.x * (blockDim.x >> 5) + waveId;
  const int nTiles = (R + 15) >> 4;
  if (tile >= nTiles) return;                       // uniform per wave

  const int rowBase = tile * 16;
  const int laneM   = lane & 15;
  const int laneN   = lane & 15;
  const int kOff    = (lane >> 4) * 2;              // 0 for lanes 0-15, 2 for 16-31
  const int row     = rowBase + laneM;
  const int rowC    = (row < R) ? row : (R - 1);    // clamp; OOB rows never stored
  const float* Xrow = X + (long long)rowC * D;

  v8f acc[8];
#pragma unroll
  for (int n = 0; n < 8; ++n) acc[n] = (v8f)0.f;

  const int kFull = D & ~3;                         // 148 for D=150

  // ---- branch-free main loop: all K strictly in-bounds ----
  for (int k0 = 0; k0 < kFull; k0 += 4) {
    const int ka = k0 + kOff;                       // ka+1 < kFull <= D
    const v2f a = *(const v2f*)(Xrow + ka);         // 8B-aligned global_load_b64
    const float* W1a = W1 + (long long)ka * A + laneN;
#pragma unroll
    for (int n = 0; n < 8; ++n) {
      v2f b;
      b.x = W1a[(n << 4)];
      b.y = W1a[(n << 4) + A];
      acc[n] = __builtin_amdgcn_wmma_f32_16x16x4_f32(
          false, a, false, b, (short)0, acc[n], false, false);
    }
  }

  // ---- single tail step: clamped loads + value selects (no exec branches) --
  if (kFull < D) {
    const int ka = kFull + kOff;
    const int kb = ka + 1;
    const int kaC = (ka < D) ? ka : (D - 1);
    const int kbC = (kb < D) ? kb : (D - 1);
    v2f a;
    a.x = Xrow[kaC];                                // garbage where B is zeroed
    a.y = Xrow[kbC];
#pragma unroll
    for (int n = 0; n < 8; ++n) {
      const int col = (n << 4) + laneN;
      float bx = W1[(long long)kaC * A + col];
      float by = W1[(long long)kbC * A + col];
      v2f b;
      b.x = (ka < D) ? bx : 0.f;                    // v_cndmask, not a branch
      b.y = (kb < D) ? by : 0.f;
      acc[n] = __builtin_amdgcn_wmma_f32_16x16x4_f32(
          false, a, false, b, (short)0, acc[n], false, false);
    }
  }

  // Fused epilogue: +b1, tanh, *q, partial per-row sums in registers.
  float p[8];
#pragma unroll
  for (int r = 0; r < 8; ++r) p[r] = 0.f;
#pragma unroll
  for (int n = 0; n < 8; ++n) {
    const float bb = b1[(n << 4) + laneN];
    const float qq = q[(n << 4) + laneN];
#pragma unroll
    for (int r = 0; r < 8; ++r) {
      p[r] += tanhf(acc[n][r] + bb) * qq;
    }
  }
  // Reduce across the 16 lanes of each half (offsets 1..8 stay in-half).
#pragma unroll
  for (int off = 1; off < 16; off <<= 1) {
#pragma unroll
    for (int r = 0; r < 8; ++r) p[r] += __shfl_xor(p[r], off, 32);
  }
  const int half = lane >> 4;                       // 0 -> rows 0-7, 1 -> rows 8-15
  if (laneN == 0) {
#pragma unroll
    for (int r = 0; r < 8; ++r) {
      const int orow = rowBase + half * 8 + r;
      if (orow < R) w[orow] = p[r];
    }
  }
}

// ---------------------------------------------------------------------------
// Kernel 3: beta = softmax over metapath axis of mean_b w[b, m].
// ---------------------------------------------------------------------------
__global__ void beta_kernel(const float* __restrict__ w,  // [B*2], m interleaved
                            float* __restrict__ beta,     // [2]
                            int B) {
  __shared__ float s0[256];
  __shared__ float s1[256];
  float a0 = 0.f, a1 = 0.f;
  for (int i = threadIdx.x; i < B; i += blockDim.x) {
    a0 += w[2 * i];
    a1 += w[2 * i + 1];
  }
  s0[threadIdx.x] = a0;
  s1[threadIdx.x] = a1;
  __syncthreads();
  for (int off = blockDim.x >> 1; off > 0; off >>= 1) {
    if ((int)threadIdx.x < off) {
      s0[threadIdx.x] += s0[threadIdx.x + off];
      s1[threadIdx.x] += s1[threadIdx.x + off];
    }
    __syncthreads();
  }
  if (threadIdx.x == 0) {
    const float m0 = s0[0] / (float)B;
    const float m1 = s1[0] / (float)B;
    const float mx = fmaxf(m0, m1);
    const float e0 = __expf(m0 - mx);
    const float e1 = __expf(m1 - mx);
    const float inv = 1.f / (e0 + e1);
    beta[0] = e0 * inv;
    beta[1] = e1 * inv;
  }
}

// ---------------------------------------------------------------------------
// Kernel 4: out[b, d] = stacked[b,0,d]*beta[0] + stacked[b,1,d]*beta[1].
// ---------------------------------------------------------------------------
__global__ void combine_kernel(const float* __restrict__ stacked,
                               const float* __restrict__ beta,
                               float* __restrict__ outp,
                               int B, int D) {
  const int i = blockIdx.x * blockDim.x + threadIdx.x;
  const int total = B * D;
  if (i >= total) return;
  const int b = i / D;
  const int d = i - b * D;
  const long long base = (long long)b * 2 * D;
  outp[i] = stacked[base + d] * beta[0] + stacked[base + D + d] * beta[1];
}

extern "C" void kernel_launch(void* const* d_in, const int* in_sizes, int n_in,
                              void* d_out, int out_size, void* d_ws, size_t ws_size,
                              hipStream_t stream) {
  const float* feat  = (const float*)d_in[0];   // out [N, D] f32
  const int*   z     = (const int*)d_in[1];     // [N]
  const int*   batch = (const int*)d_in[2];     // [N] sorted
  const float* W1    = (const float*)d_in[3];   // [D, A]
  const float* b1    = (const float*)d_in[4];   // [A]
  const float* q     = (const float*)d_in[5];   // [A]

  const int ND = in_sizes[0];
  const int N  = in_sizes[1];
  const int D  = ND / N;                        // 150
  const int A  = in_sizes[4];                   // 128
  const int B  = out_size / D;                  // 4096
  const int R  = B * 2;                         // 8192 stacked rows

  // Scratch layout: stacked [B*2*D] | w [B*2] | beta [2]
  float* stacked = (float*)d_ws;
  float* wbuf    = stacked + (size_t)R * D;
  float* beta    = wbuf + (size_t)R;

  // 1) segment means (no atomics, coalesced single pass over 600 MB input)
  {
    int threads = ((D + 31) / 32) * 32;         // 160 for D=150
    if (threads > 1024) threads = 1024;
    seg_mean_kernel<<<B, threads, 0, stream>>>(feat, z, batch, stacked, N, D);
  }

  // 2) WMMA attention scores
  {
    const int tiles = (R + 15) / 16;            // 512
    const int wavesPerBlock = 8;
    const int blocks = (tiles + wavesPerBlock - 1) / wavesPerBlock;
    attn_score_kernel<<<blocks, 32 * wavesPerBlock, 0, stream>>>(
        stacked, W1, b1, q, wbuf, R, D, A);
  }

  // 3) softmax over metapath means
  beta_kernel<<<1, 256, 0, stream>>>(wbuf, beta, B);

  // 4) blend
  {
    const int total = B * D;
    combine_kernel<<<(total + 255) / 256, 256, 0, stream>>>(stacked, beta,
                                                            (float*)d_out, B, D);
  }
}